// GNN_26362509263438
// MI455X (gfx1250) — compile-verified
//
#include <hip/hip_runtime.h>
#include <math.h>

#define NN   100000
#define EE   1600000
#define HID  128
#define BNEPS 1e-5f

typedef __attribute__((ext_vector_type(2))) float v2f;
typedef __attribute__((ext_vector_type(8))) float v8f;
typedef int vi4 __attribute__((vector_size(16)));

#if __has_builtin(__builtin_amdgcn_global_load_async_to_lds_b128)
#define USE_ASYNC_LDS 1
#endif

// ---------------- setup kernels (run once per call) ----------------

__global__ void k_zero_u32(unsigned* __restrict__ p, int n) {
  int i = blockIdx.x * blockDim.x + threadIdx.x;
  for (; i < n; i += gridDim.x * blockDim.x) p[i] = 0u;
}

// fused: weighted in-degree accumulation + CSR histogram (one edge pass)
__global__ void k_deg_hist(const int* __restrict__ ei, const float* __restrict__ ew,
                           float* __restrict__ deg, int* __restrict__ cnt) {
  int e = blockIdx.x * blockDim.x + threadIdx.x;
  if (e < EE) {
    int d = ei[EE + e];
    atomicAdd(&deg[d], ew[e]);
    atomicAdd(&cnt[d], 1);
  }
}

__global__ void k_dinv(float* __restrict__ deg) {
  int i = blockIdx.x * blockDim.x + threadIdx.x;
  if (i < NN) deg[i] = rsqrtf(deg[i] + 1.0f);
}

// Single-workgroup chunked exclusive scan: cnt_cursor[] (counts in) ->
// row_ptr[0..NN] and cnt_cursor[] becomes the running fill cursor.
__global__ __launch_bounds__(1024)
void k_scan(int* cnt_cursor, int* row_ptr) {
  __shared__ int temp[1024];
  __shared__ int carry;
  const int tid = threadIdx.x;
  if (tid == 0) carry = 0;
  __syncthreads();
  for (int base = 0; base < NN; base += 1024) {
    int i = base + tid;
    int v = (i < NN) ? cnt_cursor[i] : 0;
    temp[tid] = v;
    __syncthreads();
    for (int off = 1; off < 1024; off <<= 1) {
      int t = (tid >= off) ? temp[tid - off] : 0;
      __syncthreads();
      temp[tid] += t;
      __syncthreads();
    }
    int excl = temp[tid] - v;
    int c = carry;
    if (i < NN) { row_ptr[i] = c + excl; cnt_cursor[i] = c + excl; }
    int total = temp[1023];
    __syncthreads();
    if (tid == 0) carry = c + total;
    __syncthreads();
  }
  if (tid == 0) row_ptr[NN] = carry;
}

__global__ void k_fill(const int* __restrict__ ei, const float* __restrict__ ew,
                       const float* __restrict__ dinv, int* __restrict__ cursor,
                       int* __restrict__ csr_src, float* __restrict__ csr_w) {
  int e = blockIdx.x * blockDim.x + threadIdx.x;
  if (e >= EE) return;
  int s = ei[e], d = ei[EE + e];
  int pos = atomicAdd(&cursor[d], 1);
  csr_src[pos] = s;
  csr_w[pos] = ew[e] * dinv[s] * dinv[d];
}

// ---------------- per-layer kernels ----------------

// hw[N,128] = h[N,K] @ W[K,128] via V_WMMA_F32_16X16X4_F32.
// Block = 256 threads = 8 waves; the 16-row A tile is staged once into LDS
// (async-to-LDS), padded to stride K+4 floats so the 16-lane strided reads
// hit banks {0,4,8,...,60}: conflict-free. Wave w owns column tile w
// (8 x 16 = 128 cols). K is compile-time so the MAC loop unrolls and the
// backend pipelines 8 WMMAs with staggered s_wait_loadcnt.
template <int K>
__global__ __launch_bounds__(256)
void k_gemm_wmma(const float* __restrict__ h, const float* __restrict__ W,
                 float* __restrict__ hw) {
  constexpr int stride = K + 4;
  constexpr int KV4 = K / 4;
  __shared__ float As[16 * stride];
  const int tid = threadIdx.x;
  const int rowTile = blockIdx.x << 4;

  // cooperative, coalesced float4 staging of the 16 x K A tile
#pragma unroll
  for (int idx = tid; idx < 4 * K; idx += 256) {
    const int r = idx / KV4;
    const int c = (idx % KV4) << 2;
    const float* g = h + (size_t)(rowTile + r) * K + c;
    float* l = &As[r * stride + c];
#ifdef USE_ASYNC_LDS
    __builtin_amdgcn_global_load_async_to_lds_b128(
        (__attribute__((address_space(1))) vi4*)g,
        (__attribute__((address_space(3))) vi4*)l, 0, 0);
#else
    *(float4*)l = *(const float4*)g;
#endif
  }
#ifdef USE_ASYNC_LDS
  asm volatile("s_wait_asynccnt 0x0" ::: "memory");
#endif
  __syncthreads();

  const int lane    = tid & 31;
  const int wave    = tid >> 5;           // 0..7 column tile
  const int colTile = wave << 4;
  const int r15   = lane & 15;
  const int hi    = lane >> 4;            // lane group 0 / 1
  const int khalf = hi << 1;              // K-sub-offset 0 or 2

  const float* arow = &As[r15 * stride + khalf];
  const float* wcol = W + colTile + r15 + (size_t)khalf * HID;
  v8f acc = {0.f, 0.f, 0.f, 0.f, 0.f, 0.f, 0.f, 0.f};
#pragma unroll 8
  for (int k = 0; k < K; k += 4) {
    v2f a, b;
    a.x = arow[k];
    a.y = arow[k + 1];
    b.x = wcol[(size_t)k * HID];
    b.y = wcol[(size_t)(k + 1) * HID];
    acc = __builtin_amdgcn_wmma_f32_16x16x4_f32(false, a, false, b,
                                                (short)0, acc, false, false);
  }
  // D layout: VGPR r -> (M = rowTile + 8*hi + r, Ncol = colTile + r15)
  float* outp = hw + (size_t)(rowTile + (hi << 3)) * HID + colTile + r15;
#pragma unroll
  for (int r = 0; r < 8; ++r) outp[(size_t)r * HID] = acc[r];
}

// One wave32 per node; lane owns float4 of the 128 features.
// out[n] = hw[n]*dinv[n]^2 + sum_{e in CSR[n]} w_e * hw[src_e] + bias
__global__ __launch_bounds__(256)
void k_aggregate(const float* __restrict__ hw, const float* __restrict__ dinv,
                 const int* __restrict__ row_ptr, const int* __restrict__ csr_src,
                 const float* __restrict__ csr_w, const float* __restrict__ bias,
                 float* __restrict__ out) {
  int node = blockIdx.x * 8 + (threadIdx.x >> 5);
  if (node >= NN) return;
  int fo = (threadIdx.x & 31) * 4;

  float di = dinv[node];
  float s2 = di * di;
  float4 a = *(const float4*)(hw + (size_t)node * HID + fo);
  float4 acc;
  acc.x = a.x * s2; acc.y = a.y * s2; acc.z = a.z * s2; acc.w = a.w * s2;

  int beg = row_ptr[node], end = row_ptr[node + 1];
  for (int j = beg; j < end; ++j) {
    int src = csr_src[j];
    float w = csr_w[j];
    if (j + 1 < end)
      __builtin_prefetch(hw + (size_t)csr_src[j + 1] * HID + fo, 0, 0);
    float4 hv = *(const float4*)(hw + (size_t)src * HID + fo);
    acc.x += w * hv.x; acc.y += w * hv.y;
    acc.z += w * hv.z; acc.w += w * hv.w;
  }
  float4 b = *(const float4*)(bias + fo);
  acc.x += b.x; acc.y += b.y; acc.z += b.z; acc.w += b.w;
  *(float4*)(out + (size_t)node * HID + fo) = acc;
}

__global__ __launch_bounds__(128)
void k_bn_reduce(const float* __restrict__ h, float* __restrict__ sums,
                 float* __restrict__ sqs) {
  int f = threadIdx.x;  // feature 0..127, coalesced across lanes
  float s = 0.f, q = 0.f;
  for (int n = blockIdx.x; n < NN; n += gridDim.x) {
    float v = h[(size_t)n * HID + f];
    s += v; q += v * v;
  }
  atomicAdd(&sums[f], s);
  atomicAdd(&sqs[f], q);
}

__global__ void k_bn_finalize(const float* __restrict__ sums,
                              const float* __restrict__ sqs,
                              const float* __restrict__ gamma,
                              const float* __restrict__ beta,
                              float* __restrict__ scale,
                              float* __restrict__ shift) {
  int f = threadIdx.x;  // 1 block, 128 threads
  const float invn = 1.0f / (float)NN;
  float mean = sums[f] * invn;
  float var  = sqs[f] * invn - mean * mean;
  float sc   = gamma[f] * rsqrtf(var + BNEPS);
  scale[f] = sc;
  shift[f] = beta[f] - mean * sc;
}

__global__ __launch_bounds__(256)
void k_bn_apply(float* __restrict__ h, const float* __restrict__ scale,
                const float* __restrict__ shift, int relu, int total) {
  int i = blockIdx.x * blockDim.x + threadIdx.x;
  if (i >= total) return;
  int f = i & (HID - 1);
  float v = h[i] * scale[f] + shift[f];
  if (relu) v = fmaxf(v, 0.f);
  h[i] = v;
}

// ---------------- host orchestration ----------------

extern "C" void kernel_launch(void* const* d_in, const int* in_sizes, int n_in,
                              void* d_out, int out_size, void* d_ws, size_t ws_size,
                              hipStream_t stream) {
  const float* x     = (const float*)d_in[0];
  const int*   ei    = (const int*)d_in[1];      // [2,E]: src then dst
  const float* ew    = (const float*)d_in[2];
  const float* W_in  = (const float*)d_in[3];
  const float* b_in  = (const float*)d_in[4];
  const float* W_mid = (const float*)d_in[5];    // [3,128,128]
  const float* b_mid = (const float*)d_in[6];    // [3,128]
  const float* W_out = (const float*)d_in[7];
  const float* b_out = (const float*)d_in[8];
  const float* gam   = (const float*)d_in[9];    // [5,128]
  const float* bet   = (const float*)d_in[10];   // [5,128]
  float* out = (float*)d_out;

  // workspace carve-up (256B aligned)
  char* ws = (char*)d_ws;
  size_t off = 0;
  auto carve = [&](size_t bytes) -> char* {
    char* p = ws + off;
    off = (off + bytes + 255) & ~(size_t)255;
    return p;
  };
  float* hbuf    = (float*)carve((size_t)NN * HID * 4);
  float* hw      = (float*)carve((size_t)NN * HID * 4);
  float* dinv    = (float*)carve((size_t)NN * 4);
  int*   row_ptr = (int*)  carve((size_t)(NN + 1) * 4);
  int*   cursor  = (int*)  carve((size_t)NN * 4);
  int*   csr_src = (int*)  carve((size_t)EE * 4);
  float* csr_w   = (float*)carve((size_t)EE * 4);
  float* sums    = (float*)carve(HID * 4);
  float* sqs     = (float*)carve(HID * 4);
  float* scale   = (float*)carve(HID * 4);
  float* shift   = (float*)carve(HID * 4);
  (void)in_sizes; (void)n_in; (void)out_size; (void)ws_size;

  const int gE = (EE + 255) / 256;   // 6250
  const int gN = (NN + 255) / 256;   // 391

  // ---- graph prep: degrees, dinv, CSR with baked norms ----
  k_zero_u32<<<gN, 256, 0, stream>>>((unsigned*)dinv, NN);
  k_zero_u32<<<gN, 256, 0, stream>>>((unsigned*)cursor, NN);
  k_deg_hist<<<gE, 256, 0, stream>>>(ei, ew, dinv, cursor);
  k_dinv<<<gN, 256, 0, stream>>>(dinv);
  k_scan<<<1, 1024, 0, stream>>>(cursor, row_ptr);
  k_fill<<<gE, 256, 0, stream>>>(ei, ew, dinv, cursor, csr_src, csr_w);

  // ---- layers ----
  const float* Ws[5] = {W_in, W_mid, W_mid + 128 * 128, W_mid + 2 * 128 * 128, W_out};
  const float* bs[5] = {b_in, b_mid, b_mid + 128,       b_mid + 2 * 128,       b_out};

  const float* hcur = x;
  const int total = NN * HID;
  const int gT = (total + 255) / 256;
  for (int l = 0; l < 5; ++l) {
    float* dst = (l == 4) ? out : hbuf;
    if (l == 0)
      k_gemm_wmma<64><<<NN / 16, 256, 0, stream>>>(hcur, Ws[l], hw);
    else
      k_gemm_wmma<128><<<NN / 16, 256, 0, stream>>>(hcur, Ws[l], hw);
    k_aggregate<<<(NN + 7) / 8, 256, 0, stream>>>(hw, dinv, row_ptr, csr_src,
                                                  csr_w, bs[l], dst);
    k_zero_u32<<<1, 256, 0, stream>>>((unsigned*)sums, 2 * HID); // sums+sqs contiguous
    k_bn_reduce<<<512, 128, 0, stream>>>(dst, sums, sqs);
    k_bn_finalize<<<1, 128, 0, stream>>>(sums, sqs, gam + l * HID, bet + l * HID,
                                         scale, shift);
    k_bn_apply<<<gT, 256, 0, stream>>>(dst, scale, shift, (l < 4) ? 1 : 0, total);
    hcur = dst;
  }
}